// DGLGraphConv_22608707846293
// MI455X (gfx1250) — compile-verified
//
#include <hip/hip_runtime.h>
#include <math.h>

// ---------------------------------------------------------------------------
// CDNA5 (gfx1250) wave32 WMMA types
// ---------------------------------------------------------------------------
typedef __attribute__((ext_vector_type(2))) float v2f;   // A/B frag of 16x16x4 f32
typedef __attribute__((ext_vector_type(8))) float v8f;   // C/D frag (16x16 f32)

#define N_NODES 50000
#define N_EDGES 800000
#define IN_F    256
#define OUT_F   128
#define RANK    64

static_assert(N_NODES % 16 == 0, "row tiles exact");
static_assert(N_EDGES % 256 == 0, "edge blocks exact");

// workspace layout (float element offsets) -- total 22.5M floats = 90 MB
#define OFF_ODEG   0u
#define OFF_IDEG   50000u
#define OFF_HSUM   100000u          //  6,400,000 floats
#define OFF_LABS   6500000u         //  3,200,000 floats
#define OFF_NCNT   9700000u         //  3,200,000 ints
#define OFF_FSUM   12900000u        //  6,400,000 floats
#define OFF_FPROD  19300000u        //  3,200,000 floats
#define ZERO_CNT   12900000u        // zero through NCNT (FSUM/FPROD fully overwritten)

// ---------------------------------------------------------------------------
// 0) zero the accumulator region (uint4 stores; 0x0 is 0.0f and 0 int)
// ---------------------------------------------------------------------------
__global__ __launch_bounds__(256) void k_zero(unsigned int* __restrict__ p, unsigned n4)
{
    unsigned i = blockIdx.x * blockDim.x + threadIdx.x;
    if (i < n4) {
        ((uint4*)p)[i] = make_uint4(0u, 0u, 0u, 0u);
    }
}

// ---------------------------------------------------------------------------
// 1) degree counting (float atomics; degrees are small integers, exact in f32)
//    grid is exact (800000 = 3125 * 256) -> no guard
// ---------------------------------------------------------------------------
__global__ __launch_bounds__(256) void k_degrees(const int* __restrict__ src,
                                                 const int* __restrict__ dst,
                                                 float* __restrict__ odeg,
                                                 float* __restrict__ ideg)
{
    int e = blockIdx.x * blockDim.x + threadIdx.x;
    atomicAdd(&odeg[src[e]], 1.0f);
    atomicAdd(&ideg[dst[e]], 1.0f);
}

// ---------------------------------------------------------------------------
// 2) node GEMMs via V_WMMA_F32_16X16X4_F32.
//    Block = 128 threads (4 wave32), 16 rows of the node matrix (3125 blocks,
//    exact coverage -> no row guards anywhere).
//    A-tile (feat row scaled by out_deg^-1/2) staged in LDS, stride 260
//    => bank(r,c) = (4r + c) % 64, conflict-free for the A-fragment pattern.
//    12 N-tiles: 8 for w1 (OUT_F=128) + 4 for w2 (RANK=64); wave w does
//    tiles {w, w+4, w+8}.
// ---------------------------------------------------------------------------
__global__ __launch_bounds__(128) void k_node_gemm(const float* __restrict__ feat,
                                                   const float* __restrict__ w1,
                                                   const float* __restrict__ w2,
                                                   const float* __restrict__ odeg,
                                                   float* __restrict__ fsum,
                                                   float* __restrict__ fprod)
{
    __shared__ float As[16][260];
    __shared__ float sc[16];

    const int m0  = blockIdx.x * 16;
    const int tid = threadIdx.x;

    if (tid < 16) {
        sc[tid] = rsqrtf(fmaxf(odeg[m0 + tid], 1.0f));
    }
    __syncthreads();

    // stage A tile: 8 threads per row, 32 contiguous cols each (float4 loads)
    {
        const int r   = tid >> 3;
        const int c0  = (tid & 7) * 32;
        const float s = sc[r];
        const float4* g = (const float4*)(feat + (size_t)(m0 + r) * IN_F + c0);
#pragma unroll
        for (int j = 0; j < 8; ++j) {
            float4 x = g[j];
            As[r][c0 + 4 * j + 0] = x.x * s;
            As[r][c0 + 4 * j + 1] = x.y * s;
            As[r][c0 + 4 * j + 2] = x.z * s;
            As[r][c0 + 4 * j + 3] = x.w * s;
        }
    }
    __syncthreads();

    const int wave  = tid >> 5;
    const int lane  = tid & 31;
    const int rsel  = lane & 15;             // A: row M, B: col N, within tile
    const int khalf = (lane < 16) ? 0 : 2;   // K split across lane halves
    const int rofs  = (lane < 16) ? 0 : 8;   // C/D: VGPR r -> rows M=r, M=r+8

    for (int t = 0; t < 3; ++t) {
        const int  tile = wave + 4 * t;
        const bool isW2 = (tile >= 8);
        const int  n0   = isW2 ? (tile - 8) * 16 : tile * 16;
        const float* B  = isW2 ? w2 : w1;
        const int  ldb  = isW2 ? RANK : OUT_F;

        // warm the B panel in GL2 (global_prefetch_b8)
        for (int k = 0; k < IN_F; k += 32)
            __builtin_prefetch(B + (size_t)k * ldb + n0, 0, 0);

        v8f acc = {};
        for (int k0 = 0; k0 < IN_F; k0 += 4) {
            v2f a;
            a.x = As[rsel][k0 + khalf];
            a.y = As[rsel][k0 + khalf + 1];
            v2f b;
            const float* Bp = B + (size_t)(k0 + khalf) * ldb + n0 + rsel;
            b.x = Bp[0];
            b.y = Bp[ldb];
            acc = __builtin_amdgcn_wmma_f32_16x16x4_f32(
                false, a, false, b, (short)0, acc, false, false);
        }

        const int colg = n0 + rsel;
        if (isW2) {
            const float bias = w2[(size_t)IN_F * RANK + colg]; // the concat-ones row
            float* op = fprod + (size_t)(m0 + rofs) * RANK + colg;
#pragma unroll
            for (int r = 0; r < 8; ++r)
                op[r * RANK] = tanhf(acc[r] + bias);   // const 256B offsets
        } else {
            float* op = fsum + (size_t)(m0 + rofs) * OUT_F + colg;
#pragma unroll
            for (int r = 0; r < 8; ++r)
                op[r * OUT_F] = acc[r];                // const 512B offsets
        }
    }
}

// ---------------------------------------------------------------------------
// 3) edge scatter: one wave32 per edge (exactly 8 waves/block, 100000 blocks,
//    no guard). Coalesced gathers of the source rows, L2 float/int atomics
//    onto the destination accumulators. Working set (~90 MB) fits in the
//    192 MB L2, so this runs at L2 atomic throughput, not HBM.
// ---------------------------------------------------------------------------
__global__ __launch_bounds__(256) void k_edge(const int* __restrict__ src,
                                              const int* __restrict__ dst,
                                              const float* __restrict__ fsum,
                                              const float* __restrict__ fprod,
                                              float* __restrict__ hsum,
                                              float* __restrict__ labs,
                                              int*   __restrict__ ncnt)
{
    const int wid  = (int)((blockIdx.x * blockDim.x + threadIdx.x) >> 5);
    const int lane = threadIdx.x & 31;

    const int s = src[wid];
    const int d = dst[wid];

    const float* fs = fsum + (size_t)s * OUT_F;
    float*       hs = hsum + (size_t)d * OUT_F;
#pragma unroll
    for (int j = 0; j < 4; ++j) {
        int c = lane + 32 * j;
        atomicAdd(&hs[c], fs[c]);
    }

    const float* fp = fprod + (size_t)s * RANK;
    float*       la = labs  + (size_t)d * RANK;
    int*         nc = ncnt  + (size_t)d * RANK;
#pragma unroll
    for (int j = 0; j < 2; ++j) {
        int c = lane + 32 * j;
        float m = fp[c];
        atomicAdd(&la[c], __logf(fabsf(m)));
        if (m < 0.0f) atomicAdd(&nc[c], 1);
    }
}

// ---------------------------------------------------------------------------
// 4) finalize: h_prod = sign * exp(log_abs) (zero if in_deg==0), staged in LDS
//    (stride 68 => bank (4r+c)%64, conflict-free), then h_prod @ v with f32
//    WMMA, add h_sum, scale by in_deg^-1/2, store output. Exact 16-row blocks.
// ---------------------------------------------------------------------------
__global__ __launch_bounds__(128) void k_final(const float* __restrict__ vmat,
                                               const float* __restrict__ hsum,
                                               const float* __restrict__ labs,
                                               const int*   __restrict__ ncnt,
                                               const float* __restrict__ ideg,
                                               float* __restrict__ out)
{
    __shared__ float Hp[16][68];
    __shared__ float sc[16];

    const int m0  = blockIdx.x * 16;
    const int tid = threadIdx.x;

    if (tid < 16) {
        sc[tid] = rsqrtf(fmaxf(ideg[m0 + tid], 1.0f));
    }
    for (int i = tid; i < 16 * RANK; i += 128) {
        int r = i >> 6, c = i & 63;
        int row = m0 + r;
        float val = 0.0f;
        float dg = ideg[row];
        if (dg > 0.0f) {
            float la = labs[(size_t)row * RANK + c];
            int   n  = ncnt[(size_t)row * RANK + c];
            float sg = (n & 1) ? -1.0f : 1.0f;
            val = sg * __expf(la);
        }
        Hp[r][c] = val;
    }
    __syncthreads();

    const int wave  = tid >> 5;
    const int lane  = tid & 31;
    const int rsel  = lane & 15;
    const int khalf = (lane < 16) ? 0 : 2;
    const int rofs  = (lane < 16) ? 0 : 8;

    for (int t = 0; t < 2; ++t) {
        const int n0 = (wave * 2 + t) * 16;

        for (int k = 0; k < RANK; k += 32)
            __builtin_prefetch(vmat + (size_t)k * OUT_F + n0, 0, 0);

        v8f acc = {};
        for (int k0 = 0; k0 < RANK; k0 += 4) {
            v2f a;
            a.x = Hp[rsel][k0 + khalf];
            a.y = Hp[rsel][k0 + khalf + 1];
            v2f b;
            const float* Bp = vmat + (size_t)(k0 + khalf) * OUT_F + n0 + rsel;
            b.x = Bp[0];
            b.y = Bp[OUT_F];
            acc = __builtin_amdgcn_wmma_f32_16x16x4_f32(
                false, a, false, b, (short)0, acc, false, false);
        }
        const int colg = n0 + rsel;
        const float* hp = hsum + (size_t)(m0 + rofs) * OUT_F + colg;
        float*       op = out  + (size_t)(m0 + rofs) * OUT_F + colg;
#pragma unroll
        for (int r = 0; r < 8; ++r)
            op[r * OUT_F] = (acc[r] + hp[r * OUT_F]) * sc[rofs + r];
    }
}

// ---------------------------------------------------------------------------
// launcher
// ---------------------------------------------------------------------------
extern "C" void kernel_launch(void* const* d_in, const int* in_sizes, int n_in,
                              void* d_out, int out_size, void* d_ws, size_t ws_size,
                              hipStream_t stream)
{
    (void)in_sizes; (void)n_in; (void)out_size; (void)ws_size;

    const float* feat = (const float*)d_in[0];
    const float* w1   = (const float*)d_in[1];
    const float* w2   = (const float*)d_in[2];
    const float* vmat = (const float*)d_in[3];
    const int*   src  = (const int*)d_in[4];
    const int*   dst  = (const int*)d_in[5];

    float* ws    = (float*)d_ws;
    float* odeg  = ws + OFF_ODEG;
    float* ideg  = ws + OFF_IDEG;
    float* hsum  = ws + OFF_HSUM;
    float* labs  = ws + OFF_LABS;
    int*   ncnt  = (int*)(ws + OFF_NCNT);
    float* fsum  = ws + OFF_FSUM;
    float* fprod = ws + OFF_FPROD;

    // 0) zero accumulators (every call: launcher must be deterministic)
    {
        unsigned n4 = ZERO_CNT / 4u;
        k_zero<<<(n4 + 255u) / 256u, 256, 0, stream>>>((unsigned int*)ws, n4);
    }
    // 1) degrees (exact grid)
    k_degrees<<<N_EDGES / 256, 256, 0, stream>>>(src, dst, odeg, ideg);
    // 2) node GEMMs (feat_sum, feat_prod) -- 3125 blocks of 16 rows
    k_node_gemm<<<N_NODES / 16, 128, 0, stream>>>(feat, w1, w2, odeg, fsum, fprod);
    // 3) edge scatter: one wave32 per edge (exact grid)
    k_edge<<<(N_EDGES * 32) / 256, 256, 0, stream>>>(src, dst, fsum, fprod,
                                                     hsum, labs, ncnt);
    // 4) finalize into d_out
    k_final<<<N_NODES / 16, 128, 0, stream>>>(vmat, hsum, labs, ncnt, ideg,
                                              (float*)d_out);
}